// GCNModelVAE_13048110646074
// MI455X (gfx1250) — compile-verified
//
#include <hip/hip_runtime.h>

// GCN-VAE forward for MI455X (gfx1250, wave32).
// Pipeline: h0 = x@W1  ->  g1 = spmm(h0)  ->  h23 = relu(g1)@[W2|W3]
//        -> (mu,logvar) = spmm(h23)  ->  z = mu  ->  adj = mu @ mu^T
// All GEMMs use full-precision V_WMMA_F32_16X16X4_F32 (output is HBM-store
// bound at 256MB for adj, so fp32 matrix cores cost nothing vs bf16).
// SpMM uses global_atomic_add_f32; all operand tables are L2-resident.

typedef __attribute__((ext_vector_type(2))) float v2f;
typedef __attribute__((ext_vector_type(8))) float v8f;

#define N_NODES 8192
#define N_EDGES 262144

__device__ __forceinline__ void atomic_add_f32(float* p, float v) {
    __hip_atomic_fetch_add(p, v, __ATOMIC_RELAXED, __HIP_MEMORY_SCOPE_AGENT);
}

__global__ void zero_kernel(float* __restrict__ p, int n) {
    int i = blockIdx.x * blockDim.x + threadIdx.x;
    int stride = gridDim.x * blockDim.x;
    for (; i < n; i += stride) p[i] = 0.0f;
}

__global__ void copy_kernel(const float* __restrict__ s, float* __restrict__ d, int n) {
    int i = blockIdx.x * blockDim.x + threadIdx.x;
    int stride = gridDim.x * blockDim.x;
    for (; i < n; i += stride) d[i] = s[i];
}

// One wave computes a 16x64 strip of C = (reluA? relu(A) : A) @ B.
// f32 WMMA 16x16x4 layouts:
//   A 16x4 : lane l (l<16) holds row M=l; VGPR0/1 = K = k0+2*half {+0,+1}
//   B 4x16 : lane holds col N=lane%16;    VGPR0/1 = K = k0+2*half {+0,+1}
//   C 16x16: VGPR i -> row i + 8*half, col = lane%16
__device__ __forceinline__ void wmma_gemm_16x64(
    const float* __restrict__ A, int lda, int m0, bool reluA,
    const float* __restrict__ B, int ldb, int bn0,
    float* __restrict__ C, int ldc, int cn0, int K)
{
    const int lane = threadIdx.x & 31;
    const int half = lane >> 4;
    const int l    = lane & 15;

    v8f c0 = {}, c1 = {}, c2 = {}, c3 = {};
    const float* arow = A + (size_t)(m0 + l) * lda;

    for (int k = 0; k < K; k += 4) {
        const int ka = k + 2 * half;
        v2f a;
        a.x = arow[ka];
        a.y = arow[ka + 1];
        if (reluA) { a.x = fmaxf(a.x, 0.0f); a.y = fmaxf(a.y, 0.0f); }

        const float* bp = B + (size_t)ka * ldb + bn0 + l;
        v2f b;
        b.x = bp[0];       b.y = bp[ldb + 0];
        c0 = __builtin_amdgcn_wmma_f32_16x16x4_f32(false, a, false, b, (short)0, c0, false, false);
        b.x = bp[16];      b.y = bp[ldb + 16];
        c1 = __builtin_amdgcn_wmma_f32_16x16x4_f32(false, a, false, b, (short)0, c1, false, false);
        b.x = bp[32];      b.y = bp[ldb + 32];
        c2 = __builtin_amdgcn_wmma_f32_16x16x4_f32(false, a, false, b, (short)0, c2, false, false);
        b.x = bp[48];      b.y = bp[ldb + 48];
        c3 = __builtin_amdgcn_wmma_f32_16x16x4_f32(false, a, false, b, (short)0, c3, false, false);
    }

#pragma unroll
    for (int i = 0; i < 8; i++) {
        size_t r = (size_t)(m0 + i + 8 * half) * ldc + cn0 + l;
        C[r +  0] = c0[i];
        C[r + 16] = c1[i];
        C[r + 32] = c2[i];
        C[r + 48] = c3[i];
    }
}

// h0 = x @ W1 : [8192,512] x [512,256]. 2048 waves, each 16x64.
__global__ void gemm1_kernel(const float* __restrict__ x,
                             const float* __restrict__ W1,
                             float* __restrict__ h0)
{
    int wave = (blockIdx.x * blockDim.x + threadIdx.x) >> 5;
    int m0 = (wave >> 2) << 4;      // 512 m-tiles
    int n0 = (wave & 3) << 6;       // 4 n-blocks of 64
    wmma_gemm_16x64(x, 512, m0, false, W1, 256, n0, h0, 256, n0, 512);
}

// h23 = relu(g1) @ [W2 | W3] : [8192,256] x [256,(128|128)]. 2048 waves.
__global__ void gemm2_kernel(const float* __restrict__ g1,
                             const float* __restrict__ W2,
                             const float* __restrict__ W3,
                             float* __restrict__ h23)
{
    int wave = (blockIdx.x * blockDim.x + threadIdx.x) >> 5;
    int m0 = (wave >> 2) << 4;
    int nb = wave & 3;                           // 64-wide column block of output
    const float* B = (nb < 2) ? W2 : W3;
    int bn0 = (nb & 1) << 6;                     // column inside W2/W3
    wmma_gemm_16x64(g1, 256, m0, true, B, 128, bn0, h23, 256, nb << 6, 256);
}

// spmm layer 1: g1[dst] += w * h0[src], D = 256. One wave per edge.
__global__ void spmm1_kernel(const int* __restrict__ src,
                             const int* __restrict__ dst,
                             const float* __restrict__ wgt,
                             const float* __restrict__ h,
                             float* __restrict__ out)
{
    int edge = (blockIdx.x * blockDim.x + threadIdx.x) >> 5;
    int lane = threadIdx.x & 31;
    int s = src[edge], d = dst[edge];
    float w = wgt[edge];
    const float* hr = h + (size_t)s * 256;
    float* orow = out + (size_t)d * 256;
#pragma unroll
    for (int i = 0; i < 8; i++) {
        int col = lane + 32 * i;
        atomic_add_f32(&orow[col], w * hr[col]);
    }
}

// spmm layers 2+3 fused: cols [0,128) -> mu, [128,256) -> logvar.
__global__ void spmm23_kernel(const int* __restrict__ src,
                              const int* __restrict__ dst,
                              const float* __restrict__ wgt,
                              const float* __restrict__ h23,
                              float* __restrict__ mu,
                              float* __restrict__ logvar)
{
    int edge = (blockIdx.x * blockDim.x + threadIdx.x) >> 5;
    int lane = threadIdx.x & 31;
    int s = src[edge], d = dst[edge];
    float w = wgt[edge];
    const float* hr = h23 + (size_t)s * 256;
#pragma unroll
    for (int i = 0; i < 8; i++) {
        int col = lane + 32 * i;
        float v = w * hr[col];
        float* t = (col < 128) ? (mu + (size_t)d * 128 + col)
                               : (logvar + (size_t)d * 128 + (col - 128));
        atomic_add_f32(t, v);
    }
}

// adj = Z @ Z^T : [8192,128] x [128,8192]. Wave: 16x64 strip; block: 16x512.
// Both A and B fragments are contiguous K-pairs of a Z row -> global_load_b64.
__global__ void adj_kernel(const float* __restrict__ Z, float* __restrict__ adj)
{
    const int wave = threadIdx.x >> 5;
    const int lane = threadIdx.x & 31;
    const int half = lane >> 4;
    const int l    = lane & 15;

    const int m0 = blockIdx.y << 4;                     // 512 m-tiles
    const int n0 = (blockIdx.x * 8 + wave) << 6;        // 128 n-strips of 64

    const float* ar  = Z + (size_t)(m0 + l) * 128;
    const float* br0 = Z + (size_t)(n0 +  0 + l) * 128;
    const float* br1 = Z + (size_t)(n0 + 16 + l) * 128;
    const float* br2 = Z + (size_t)(n0 + 32 + l) * 128;
    const float* br3 = Z + (size_t)(n0 + 48 + l) * 128;

    v8f c0 = {}, c1 = {}, c2 = {}, c3 = {};
#pragma unroll 4
    for (int k = 0; k < 128; k += 4) {
        const int ka = k + 2 * half;
        v2f a  = *(const v2f*)(ar  + ka);
        v2f b0 = *(const v2f*)(br0 + ka);
        v2f b1 = *(const v2f*)(br1 + ka);
        v2f b2 = *(const v2f*)(br2 + ka);
        v2f b3 = *(const v2f*)(br3 + ka);
        c0 = __builtin_amdgcn_wmma_f32_16x16x4_f32(false, a, false, b0, (short)0, c0, false, false);
        c1 = __builtin_amdgcn_wmma_f32_16x16x4_f32(false, a, false, b1, (short)0, c1, false, false);
        c2 = __builtin_amdgcn_wmma_f32_16x16x4_f32(false, a, false, b2, (short)0, c2, false, false);
        c3 = __builtin_amdgcn_wmma_f32_16x16x4_f32(false, a, false, b3, (short)0, c3, false, false);
    }

#pragma unroll
    for (int i = 0; i < 8; i++) {
        size_t r = (size_t)(m0 + i + 8 * half) * 8192 + n0 + l;
        adj[r +  0] = c0[i];
        adj[r + 16] = c1[i];
        adj[r + 32] = c2[i];
        adj[r + 48] = c3[i];
    }
}

extern "C" void kernel_launch(void* const* d_in, const int* in_sizes, int n_in,
                              void* d_out, int out_size, void* d_ws, size_t ws_size,
                              hipStream_t stream) {
    const float* x   = (const float*)d_in[0];
    const int*   ei  = (const int*)d_in[1];    // [2, E]
    const float* ew  = (const float*)d_in[2];
    const float* W1  = (const float*)d_in[3];
    const float* W2  = (const float*)d_in[4];
    const float* W3  = (const float*)d_in[5];
    float* out = (float*)d_out;

    const int* src = ei;
    const int* dst = ei + N_EDGES;

    // workspace: h0 [8192x256], g1 [8192x256], h23 [8192x256] = 24 MB
    float* h0  = (float*)d_ws;
    float* g1  = h0 + (size_t)N_NODES * 256;
    float* h23 = g1 + (size_t)N_NODES * 256;

    // outputs: adj [8192x8192], mu [8192x128], logvar [8192x128], z [8192x128]
    float* adj    = out;
    float* mu     = out + (size_t)N_NODES * N_NODES;
    float* logvar = mu  + (size_t)N_NODES * 128;
    float* z      = logvar + (size_t)N_NODES * 128;

    const int nAccum = N_NODES * 256;   // 2M floats

    zero_kernel<<<2048, 256, 0, stream>>>(g1, nAccum);
    zero_kernel<<<2048, 256, 0, stream>>>(mu, nAccum);        // mu+logvar contiguous

    gemm1_kernel<<<256, 256, 0, stream>>>(x, W1, h0);          // 2048 waves
    spmm1_kernel<<<N_EDGES / 8, 256, 0, stream>>>(src, dst, ew, h0, g1);
    gemm2_kernel<<<256, 256, 0, stream>>>(g1, W2, W3, h23);    // 2048 waves
    spmm23_kernel<<<N_EDGES / 8, 256, 0, stream>>>(src, dst, ew, h23, mu, logvar);
    copy_kernel<<<1024, 256, 0, stream>>>(mu, z, N_NODES * 128);
    adj_kernel<<<dim3(16, 512), 256, 0, stream>>>(mu, adj);    // 128x512 wave strips
}